// ChamferEigenRatioLoss_28484223107626
// MI455X (gfx1250) — compile-verified
//
#include <hip/hip_runtime.h>

// ---------------------------------------------------------------------------
// ChamferEigenRatioLoss for MI455X (gfx1250, wave32, WMMA)
//
// Core op everywhere: distance surrogate via V_WMMA_F32_16X16X4_F32
//   A (16x4, candidates m): K = (cx, cy, cz, |c|^2)
//   B (4x16, queries n):    K = (-2qx, -2qy, -2qz, 1)
//   D[m][n] = |c_m|^2 - 2 q_n . c_m   (== sqdist minus per-query constant)
// ---------------------------------------------------------------------------

typedef __attribute__((ext_vector_type(2))) float v2f;
typedef __attribute__((ext_vector_type(8))) float v8f;

#define CHUNK 2048          // candidate points staged in LDS per pass (32 KB)
#define BDIM  256           // 8 waves
#define QPB   128           // queries per block = 8 waves * 16 columns

// Build A operand from LDS-staged packed float4 candidates and run the WMMA.
// ISA A layout (32-bit 16x4): lanes 0-15 hold M=lane with K=0,1 in 2 VGPRs,
// lanes 16-31 hold M=lane-16 with K=2,3.  Packed point = (x,y,z,|p|^2), so
// lane<16 reads floats [0,1], lane>=16 reads floats [2,3]: one ds_load_b64.
__device__ __forceinline__ v8f tile_dist(const float* scf, int t, int lane, v2f bop) {
  const int mloc = t * 16 + (lane & 15);
  const float2 av = *(const float2*)(scf + mloc * 4 + ((lane >> 4) << 1));
  v2f a;
  a[0] = av.x;
  a[1] = av.y;
  v8f cacc = {};
  return __builtin_amdgcn_wmma_f32_16x16x4_f32(false, a, false, bop, (short)0, cacc,
                                               false, false);
}

// B operand (4x16): lanes 0-15: K=0,1 for column n=lane; lanes 16-31: K=2,3.
__device__ __forceinline__ v2f make_bop(float4 qp, int lane) {
  v2f b;
  b[0] = (lane < 16) ? -2.0f * qp.x : -2.0f * qp.z;
  b[1] = (lane < 16) ? -2.0f * qp.y : 1.0f;
  return b;
}

// --------------------------- pack: (x,y,z) -> (x,y,z,|p|^2) -----------------
__global__ void pack_kernel(const float* __restrict__ pts, float4* __restrict__ out,
                            int total) {
  int i = blockIdx.x * blockDim.x + threadIdx.x;
  if (i < total) {
    float a = pts[3 * i + 0];
    float b = pts[3 * i + 1];
    float c = pts[3 * i + 2];
    out[i] = make_float4(a, b, c, a * a + b * b + c * c);
  }
}

// --------------------------- cross argmin (Chamfer NN) ----------------------
__global__ __launch_bounds__(BDIM) void nn_kernel(const float4* __restrict__ qpts,
                                                  const float4* __restrict__ cpts,
                                                  int* __restrict__ out_idx, int n) {
  __shared__ float4 sc[CHUNK];
  const int b = blockIdx.y;
  const float4* q = qpts + (size_t)b * n;
  const float4* c = cpts + (size_t)b * n;
  int* oi = out_idx + (size_t)b * n;

  const int tid  = threadIdx.x;
  const int lane = tid & 31;
  const int wave = tid >> 5;
  const int qn   = blockIdx.x * QPB + wave * 16 + (lane & 15);

  const v2f bop = make_bop(q[qn], lane);

  float best    = 3.4e38f;
  int   bestIdx = 0;

  for (int ch = 0; ch < n; ch += CHUNK) {
    __syncthreads();
    for (int i = tid; i < CHUNK; i += BDIM) sc[i] = c[ch + i];
    if (ch + CHUNK < n) __builtin_prefetch(&c[ch + CHUNK + tid * 8], 0, 0);
    __syncthreads();
    const float* scf = (const float*)sc;
    for (int t = 0; t < CHUNK / 16; ++t) {
      v8f d = tile_dist(scf, t, lane, bop);
      const int base = ch + t * 16 + ((lane >> 4) << 3);
#pragma unroll
      for (int r = 0; r < 8; ++r) {
        float dv = d[r];
        if (dv < best) { best = dv; bestIdx = base + r; }
      }
    }
  }
  // combine row halves 0-7 / 8-15 across the lane pair (L, L+16)
  float ob   = __shfl_xor(best, 16);
  int   oidx = __shfl_xor(bestIdx, 16);
  if (ob < best || (ob == best && oidx < bestIdx)) { best = ob; bestIdx = oidx; }
  if (lane < 16) oi[qn] = bestIdx;
}

// --------------------------- eigen-ratio (K=16 NN + 3x3 eig) ----------------
__global__ __launch_bounds__(BDIM) void eigen_kernel(const float4* __restrict__ pts,
                                                     float* __restrict__ er, int n) {
  __shared__ float4 sc[CHUNK];
  const int b = blockIdx.y;
  const float4* c = pts + (size_t)b * n;
  float* ero = er + (size_t)b * n;

  const int tid  = threadIdx.x;
  const int lane = tid & 31;
  const int wave = tid >> 5;
  const int qn   = blockIdx.x * QPB + wave * 16 + (lane & 15);

  const v2f bop = make_bop(c[qn], lane);

  float nd[16];
  int   ni[16];
#pragma unroll
  for (int j = 0; j < 16; ++j) { nd[j] = 3.4e38f; ni[j] = 0; }

  for (int ch = 0; ch < n; ch += CHUNK) {
    __syncthreads();
    for (int i = tid; i < CHUNK; i += BDIM) sc[i] = c[ch + i];
    if (ch + CHUNK < n) __builtin_prefetch(&c[ch + CHUNK + tid * 8], 0, 0);
    __syncthreads();
    const float* scf = (const float*)sc;
    for (int t = 0; t < CHUNK / 16; ++t) {
      v8f d = tile_dist(scf, t, lane, bop);
      const int base = ch + t * 16 + ((lane >> 4) << 3);
#pragma unroll
      for (int r = 0; r < 8; ++r) {
        float dv = d[r];
        if (dv < nd[15]) {  // sorted ascending; nd[15] is the current 16th-smallest
          nd[15] = dv;
          ni[15] = base + r;
#pragma unroll
          for (int s = 15; s > 0; --s) {
            if (nd[s] < nd[s - 1]) {
              float td = nd[s]; nd[s] = nd[s - 1]; nd[s - 1] = td;
              int   ti = ni[s]; ni[s] = ni[s - 1]; ni[s - 1] = ti;
            }
          }
        }
      }
    }
  }

  // Bitonic pairwise-min merge of two sorted 16-lists across lanes (L, L+16):
  // top-16 of union = elementwise min(a[i], b[15-i]).  Snapshot partner first.
  float pd[16];
  int   pi[16];
#pragma unroll
  for (int j = 0; j < 16; ++j) {
    pd[j] = __shfl_xor(nd[j], 16);
    pi[j] = __shfl_xor(ni[j], 16);
  }
#pragma unroll
  for (int j = 0; j < 16; ++j) {
    float bd = pd[15 - j];
    if (bd < nd[j]) { nd[j] = bd; ni[j] = pi[15 - j]; }
  }

  if (lane < 16) {
    // covariance of the 16 neighbors: E[pp^T] - mu mu^T  (gathers hit L2/WGP$)
    float sx = 0, sy = 0, sz = 0;
    float sxx = 0, sxy = 0, sxz = 0, syy = 0, syz = 0, szz = 0;
#pragma unroll
    for (int j = 0; j < 16; ++j) {
      float4 p = c[ni[j]];
      sx += p.x; sy += p.y; sz += p.z;
      sxx += p.x * p.x; sxy += p.x * p.y; sxz += p.x * p.z;
      syy += p.y * p.y; syz += p.y * p.z; szz += p.z * p.z;
    }
    const float invK = 1.0f / 16.0f;
    float mx = sx * invK, my = sy * invK, mz = sz * invK;
    float cxx = sxx * invK - mx * mx;
    float cxy = sxy * invK - mx * my;
    float cxz = sxz * invK - mx * mz;
    float cyy = syy * invK - my * my;
    float cyz = syz * invK - my * mz;
    float czz = szz * invK - mz * mz;

    // closed-form symmetric 3x3 eigenvalues
    float qd = (cxx + cyy + czz) * (1.0f / 3.0f);
    float p1 = cxy * cxy + cxz * cxz + cyz * cyz;
    float axx = cxx - qd, ayy = cyy - qd, azz = czz - qd;
    float p2 = axx * axx + ayy * ayy + azz * azz + 2.0f * p1;
    float p = sqrtf(p2 * (1.0f / 6.0f));
    float ratio;
    if (p < 1e-12f) {
      ratio = 1.0f;  // all eigenvalues equal
    } else {
      float ip = 1.0f / p;
      float bxx = axx * ip, byy = ayy * ip, bzz = azz * ip;
      float bxy = cxy * ip, bxz = cxz * ip, byz = cyz * ip;
      float detB = bxx * (byy * bzz - byz * byz) - bxy * (bxy * bzz - byz * bxz) +
                   bxz * (bxy * byz - byy * bxz);
      float r = 0.5f * detB;
      r = fminf(1.0f, fmaxf(-1.0f, r));
      float phi = acosf(r) * (1.0f / 3.0f);
      float lmax = qd + 2.0f * p * __cosf(phi);
      float lmin = qd + 2.0f * p * __cosf(phi + 2.0943951023931953f);
      float lmid = 3.0f * qd - lmax - lmin;
      ratio = lmax / lmid;
    }
    ero[qn] = ratio;
  }
}

// --------------------------- final scalar loss ------------------------------
__global__ void loss_kernel(const float* __restrict__ er1, const float* __restrict__ er2,
                            const int* __restrict__ idx1, const int* __restrict__ idx2,
                            float* __restrict__ out, int n) {
  __shared__ float red[256];
  const int tid = threadIdx.x;
  float acc = 0.0f;
  for (int t = tid; t < 2 * n; t += 256) {
    int b = t / n;
    int i = t - b * n;
    int base = b * n;
    float a = er1[base + i] - er2[base + idx1[base + i]];
    float c = er2[base + i] - er1[base + idx2[base + i]];
    acc += a * a + c * c;
  }
  red[tid] = acc;
  __syncthreads();
  for (int s = 128; s > 0; s >>= 1) {
    if (tid < s) red[tid] += red[tid + s];
    __syncthreads();
  }
  // loss = mean_b 0.5*(er_d1 + er_d2), er_d = mean_n(...)  ->  sum * 0.5/(B*n)
  if (tid == 0) out[0] = red[0] * (0.5f / (2.0f * (float)n));
}

// ---------------------------------------------------------------------------
extern "C" void kernel_launch(void* const* d_in, const int* in_sizes, int n_in,
                              void* d_out, int out_size, void* d_ws, size_t ws_size,
                              hipStream_t stream) {
  (void)n_in; (void)out_size; (void)ws_size;
  const float* x = (const float*)d_in[0];
  const float* y = (const float*)d_in[1];
  const int B = 2;
  const int n = in_sizes[0] / (3 * B);  // 8192
  const int total = B * n;
  float* out = (float*)d_out;

  // workspace layout (~768 KB)
  float4* px = (float4*)d_ws;
  float4* py = px + total;
  int* idx1  = (int*)(py + total);
  int* idx2  = idx1 + total;
  float* er1 = (float*)(idx2 + total);
  float* er2 = er1 + total;

  pack_kernel<<<(total + 255) / 256, 256, 0, stream>>>(x, px, total);
  pack_kernel<<<(total + 255) / 256, 256, 0, stream>>>(y, py, total);

  dim3 grid(n / QPB, B);
  nn_kernel<<<grid, BDIM, 0, stream>>>(px, py, idx1, n);   // x -> nearest y
  nn_kernel<<<grid, BDIM, 0, stream>>>(py, px, idx2, n);   // y -> nearest x
  eigen_kernel<<<grid, BDIM, 0, stream>>>(px, er1, n);
  eigen_kernel<<<grid, BDIM, 0, stream>>>(py, er2, n);

  loss_kernel<<<1, 256, 0, stream>>>(er1, er2, idx1, idx2, out, n);
}